// QuantumWalk_31877247271333
// MI455X (gfx1250) — compile-verified
//
#include <hip/hip_runtime.h>

// ---------------------------------------------------------------------------
// Quantum walk, fully fused for MI455X (gfx1250, wave32).
//   B=8, N=2048, D=8, C=128, F=256, T=4
// One workgroup: 128 n-rows (+16 halo) x 8 d x 16 channels, all T steps in
// LDS (2 x 80KB ping-pong). Tile load uses GLOBAL_LOAD_ASYNC_TO_LDS_B128
// (ASYNCcnt) overlapped with swap-index preprocessing. Final reduction GEMM
// d^T(16x128) @ neibs(128x256) uses V_WMMA_F32_16X16X4_F32, accumulated with
// f32 global atomics into a pre-zeroed d_out.
// ---------------------------------------------------------------------------

namespace {
constexpr int Bc = 8;
constexpr int Nc = 2048;
constexpr int Dc = 8;
constexpr int Cc = 128;
constexpr int Fc = 256;
constexpr int Tc = 4;

constexpr int ROWS  = 128;            // output rows per tile
constexpr int HALO  = 4 * Tc;         // gather radius 4 per step
constexpr int TROWS = ROWS + 2*HALO;  // 160
constexpr int CCH   = 16;             // channels per workgroup == WMMA M
constexpr int NTHR  = 256;            // 8 wave32s
}

typedef __attribute__((ext_vector_type(2))) float v2f;
typedef __attribute__((ext_vector_type(8))) float v8f;
typedef __attribute__((__vector_size__(4 * sizeof(int)))) int v4i;

typedef __attribute__((address_space(1))) v4i* gv4i_p;  // global int4*
typedef __attribute__((address_space(3))) v4i* lv4i_p;  // LDS int4*

#if __has_builtin(__builtin_amdgcn_global_load_async_to_lds_b128)
#define HAS_ASYNC_LDS 1
#else
#define HAS_ASYNC_LDS 0
#endif

__global__ void qw_zero(float* p, int n) {
  int i = blockIdx.x * blockDim.x + threadIdx.x;
  if (i < n) p[i] = 0.0f;
}

__global__ __launch_bounds__(NTHR) void qw_fused(
    const float* __restrict__ amps,    // [B,N,D,C]
    const float* __restrict__ neibs,   // [B,N,F]
    const float* __restrict__ coins,   // [T,D,D]
    const int*   __restrict__ swap_a,  // [B,N,D]
    const int*   __restrict__ swap_b,  // [B,N,D]
    float*       __restrict__ out)     // [B*C, F]
{
  __shared__ float bufA[TROWS * Dc * CCH];   // 80 KB
  __shared__ float bufB[TROWS * Dc * CCH];   // 80 KB
  __shared__ float dsq[ROWS * CCH];          //  8 KB
  __shared__ int   srcoff[TROWS * Dc];       //  5 KB
  __shared__ float cmat[Dc * Dc];

  const int tid    = threadIdx.x;
  const int cblk   = blockIdx.x & 7;          // C / CCH = 8
  const int rowblk = (blockIdx.x >> 3) & 15;  // N / ROWS = 16
  const int b      = blockIdx.x >> 7;         // B = 8
  const int c0     = cblk * CCH;
  const int j0     = rowblk * ROWS;

  // ---- tile load: [j0-HALO, j0+ROWS+HALO) x D x [c0,c0+CCH) -> bufA -------
  // Async direct-to-LDS (ASYNCcnt); overlapped with the srcoff precompute.
  for (int idx = tid; idx < TROWS * Dc * (CCH/4); idx += NTHR) {
    int c4 = idx % (CCH/4);
    int r  = idx / (CCH/4);
    int d  = r % Dc;
    int lj = r / Dc;
    int jg = (j0 - HALO + lj + Nc) & (Nc - 1);
    const float* src = &amps[(((size_t)b*Nc + jg)*Dc + d)*Cc + c0 + 4*c4];
    float*       dst = &bufA[(lj*Dc + d)*CCH + 4*c4];
#if HAS_ASYNC_LDS
    __builtin_amdgcn_global_load_async_to_lds_b128(
        (gv4i_p)src, (lv4i_p)dst, /*offset=*/0, /*cpol=*/0);
#else
    *(float4*)dst = *(const float4*)src;
#endif
  }

  // ---- precompute tile-local gather offsets (swap indices are t-invariant) -
  for (int idx = tid; idx < TROWS * Dc; idx += NTHR) {
    int k  = idx % Dc;
    int lj = idx / Dc;
    int jg = (j0 - HALO + lj + Nc) & (Nc - 1);
    size_t gi = ((size_t)b*Nc + jg)*Dc + k;
    int sa = swap_a[gi];
    int sb = swap_b[gi];
    int delta = sa - jg;
    if (delta >= Nc/2)  delta -= Nc;   // circulant wrap -> [-4,4]
    if (delta < -Nc/2)  delta += Nc;
    srcoff[idx] = (lj + delta) * Dc + sb;
  }

#if HAS_ASYNC_LDS
#if __has_builtin(__builtin_amdgcn_s_wait_asynccnt)
  __builtin_amdgcn_s_wait_asynccnt(0);
#else
  asm volatile("s_wait_asynccnt 0x0" ::: "memory");
#endif
#endif
  __syncthreads();

  float* cur = bufA;
  float* nxt = bufB;

  // ---- T coin+shift steps, valid region shrinks by 4 rows/side per step ----
  for (int t = 0; t < Tc; ++t) {
    if (tid < Dc*Dc) cmat[tid] = coins[t*Dc*Dc + tid];
    __syncthreads();

    // coin: out[e] = sum_d in[d] * coin[d][e]   (in-place, thread-private)
    {
      const int lo = HALO - 4*(Tc - t);
      const int nrows = ROWS + 8*(Tc - t);
      for (int idx = tid; idx < nrows * CCH; idx += NTHR) {
        int c  = idx % CCH;
        int lj = lo + idx / CCH;
        float in[Dc], o[Dc];
        #pragma unroll
        for (int d = 0; d < Dc; ++d) in[d] = cur[(lj*Dc + d)*CCH + c];
        #pragma unroll
        for (int e = 0; e < Dc; ++e) {
          float s = 0.0f;
          #pragma unroll
          for (int d = 0; d < Dc; ++d) s += in[d] * cmat[d*Dc + e];
          o[e] = s;
        }
        #pragma unroll
        for (int e = 0; e < Dc; ++e) cur[(lj*Dc + e)*CCH + c] = o[e];
      }
    }
    __syncthreads();

    // shift: nxt[lj,k] = cur[srcoff[lj,k]]
    {
      const int lo = HALO - 4*(Tc - t - 1);
      const int nrows = ROWS + 8*(Tc - t - 1);
      for (int idx = tid; idx < nrows * Dc * CCH; idx += NTHR) {
        int c  = idx % CCH;
        int r  = idx / CCH;           // (lj-lo)*D + k
        int k  = r % Dc;
        int lj = lo + r / Dc;
        int so = srcoff[lj*Dc + k];
        nxt[(lj*Dc + k)*CCH + c] = cur[so*CCH + c];
      }
    }
    __syncthreads();
    float* tmp = cur; cur = nxt; nxt = tmp;
  }

  // ---- d[n,c] = sum_d amps^2  (rows HALO..HALO+ROWS) ----
  for (int idx = tid; idx < ROWS * CCH; idx += NTHR) {
    int c = idx % CCH;
    int r = idx / CCH;
    int lj = HALO + r;
    float s = 0.0f;
    #pragma unroll
    for (int d = 0; d < Dc; ++d) {
      float v = cur[(lj*Dc + d)*CCH + c];
      s += v * v;
    }
    dsq[r*CCH + c] = s;
  }
  __syncthreads();

  // ---- out[c0+m, f] += dsq^T(16 x 128) @ neibs(128 x 256) via WMMA f32 -----
  // V_WMMA_F32_16X16X4_F32: A 16x4 (M=c, K=n), B 4x16 (K=n, N=f), C/D 16x16.
  const int wave  = tid >> 5;
  const int lane  = tid & 31;
  const int laneM = lane & 15;
  const bool hi   = lane >= 16;

  for (int ftile = wave; ftile < Fc/16; ftile += NTHR/32) {
    const int f0 = ftile * 16;
    v8f acc = {};
    for (int n0 = 0; n0 < ROWS; n0 += 4) {
      const int kb = n0 + (hi ? 2 : 0);
      v2f a, bb;
      // A: lane m holds A[m][K]; VGPR0 = K={0|2}, VGPR1 = K={1|3}
      a[0] = dsq[(kb + 0)*CCH + laneM];
      a[1] = dsq[(kb + 1)*CCH + laneM];
      // B: lane n holds B[K][n]; rows j0+n0.. are in [0,N)
      const float* nb = neibs + ((size_t)b*Nc + (j0 + kb))*Fc + f0 + laneM;
      bb[0] = nb[0];
      bb[1] = nb[Fc];
      acc = __builtin_amdgcn_wmma_f32_16x16x4_f32(
          /*neg_a=*/false, a, /*neg_b=*/false, bb,
          /*c_mod=*/(short)0, acc, /*reuse_a=*/false, /*reuse_b=*/false);
    }
    // D layout: VGPR i -> (M=i, N=lane) for lanes 0-15, (M=i+8, N=lane-16) hi
    const int mbase = hi ? 8 : 0;
    #pragma unroll
    for (int i = 0; i < 8; ++i) {
      const int cg = c0 + mbase + i;
      atomicAdd(&out[((size_t)b*Cc + cg)*Fc + f0 + laneM], acc[i]);
    }
  }
}

extern "C" void kernel_launch(void* const* d_in, const int* in_sizes, int n_in,
                              void* d_out, int out_size, void* d_ws, size_t ws_size,
                              hipStream_t stream) {
  const float* amps   = (const float*)d_in[0];
  const float* neibs  = (const float*)d_in[1];
  const float* coins  = (const float*)d_in[2];
  const int*   swap_a = (const int*)d_in[3];
  const int*   swap_b = (const int*)d_in[4];
  float*       out    = (float*)d_out;

  const int outN = Bc * Cc * Fc; // 262144
  qw_zero<<<(outN + 255) / 256, 256, 0, stream>>>(out, outN);

  const int grid = Bc * (Nc / ROWS) * (Cc / CCH); // 8*16*8 = 1024
  qw_fused<<<grid, NTHR, 0, stream>>>(amps, neibs, coins, swap_a, swap_b, out);
}